// GRUD_22187801051696
// MI455X (gfx1250) — compile-verified
//
#include <hip/hip_runtime.h>
#include <hip/hip_bf16.h>

#define BATCH 128
#define FEAT  256
#define TLEN  2048
#define NOUT  2
#define TT    8              // time tile per TDM transfer
#define RSTR  (TT + 1)       // padded LDS row stride (floats): 9, coprime with 64 banks

typedef __attribute__((ext_vector_type(2))) float v2f;
typedef __attribute__((ext_vector_type(8))) float v8f;
typedef __attribute__((ext_vector_type(4))) unsigned int u32x4;
typedef __attribute__((ext_vector_type(4))) int i32x4;
typedef __attribute__((ext_vector_type(8))) int i32x8;

// ---- fast transcendentals (lower to v_exp_f32 / v_rcp_f32) ----
__device__ __forceinline__ float fexp(float x) {
    return __builtin_amdgcn_exp2f(x * 1.44269504088896340736f);
}
__device__ __forceinline__ float fsigmoid(float x) {
    return __builtin_amdgcn_rcpf(1.0f + fexp(-x));
}
__device__ __forceinline__ float ftanh(float x) {
    return __builtin_fmaf(2.0f, fsigmoid(2.0f * x), -1.0f);
}

// ============================================================
// TDM: issue one 2D tensor_load_to_lds.
//   tile = [TT elements along T] x [256 chain rows]
//   global row stride = TLEN elements; LDS row stride = TT+1 DWORDs
//   (pad_enable, pad_interval = 8 DWORDs -> code 2, pad_amount = 1 DWORD -> code 0)
// ============================================================
__device__ __forceinline__ void tdm_load_2d(unsigned lds_byte_addr, const float* gptr) {
    const unsigned long long ga = (unsigned long long)(uintptr_t)gptr;

    u32x4 g0;
    g0[0] = 1u;                                        // count=1 valid user D#
    g0[1] = lds_byte_addr;                             // LDS dest address
    g0[2] = (unsigned)(ga & 0xFFFFFFFFu);              // global_addr[31:0]
    g0[3] = (unsigned)((ga >> 32) & 0x01FFFFFFu)       // global_addr[56:32]
          | (2u << 30);                                // type = 2 ("image")

    i32x8 g1;
    g1[0] = (int)((2u << 16)                           // data_size = 4 bytes
                | (1u << 20)                           // pad_enable
                | (2u << 22)                           // pad_interval: 8 DWORDs
                | (0u << 25));                         // pad_amount: 1 DWORD
    g1[1] = (int)(((unsigned)TLEN & 0xFFFFu) << 16);   // tensor_dim0[15:0] (bits 63:48)
    g1[2] = (int)(((unsigned)TLEN >> 16)               // tensor_dim0[31:16]
                | (((unsigned)(BATCH * FEAT)) << 16)); // tensor_dim1[15:0]
    g1[3] = (int)((((unsigned)(BATCH * FEAT)) >> 16)   // tensor_dim1[31:16]
                | ((unsigned)TT << 16));               // tile_dim0 = TT
    g1[4] = (int)256u;                                 // tile_dim1 = 256 rows; tile_dim2 = 0
    g1[5] = (int)TLEN;                                 // tensor_dim0_stride[31:0]
    g1[6] = 0;                                         // stride hi | dim1_stride lo
    g1[7] = 0;

    const i32x4 z4 = {0, 0, 0, 0};
#if defined(__clang_major__) && (__clang_major__ >= 23)
    const i32x8 z8 = {0, 0, 0, 0, 0, 0, 0, 0};
    __builtin_amdgcn_tensor_load_to_lds(g0, g1, z4, z4, z8, 0);
#else
    __builtin_amdgcn_tensor_load_to_lds(g0, g1, z4, z4, 0);
#endif
}

// ============================================================
// Kernel 1: GRU-D recurrence. Block = 256 threads = 256 chains.
// Double-buffered TDM staging: the Tensor Data Mover gathers the
// strided [256 x TT] tile (global row stride 8KB) into LDS with a
// bank-conflict-free padded layout, while the VALU works on the
// other buffer. Recurrence itself stays in registers.
// ============================================================
__global__ __launch_bounds__(256) void grud_scan_kernel(
    const float* __restrict__ X, const float* __restrict__ Mask,
    const float* __restrict__ Delta,
    const float* __restrict__ x_mean,
    const float* __restrict__ w_dg_x, const float* __restrict__ w_dg_h,
    const float* __restrict__ w_xz, const float* __restrict__ w_hz,
    const float* __restrict__ w_mz,
    const float* __restrict__ w_xr, const float* __restrict__ w_hr,
    const float* __restrict__ w_mr,
    const float* __restrict__ w_xh, const float* __restrict__ w_hh,
    const float* __restrict__ w_mh,
    const float* __restrict__ b_dg_x, const float* __restrict__ b_dg_h,
    const float* __restrict__ b_z, const float* __restrict__ b_r,
    const float* __restrict__ b_h,
    float* __restrict__ hout)
{
    __shared__ float sX[2][256 * RSTR];   // 3 arrays x 2 buffers x 9216B = 54KB
    __shared__ float sM[2][256 * RSTR];
    __shared__ float sD[2][256 * RSTR];

    const int tidb   = threadIdx.x;              // chain within block
    const int chain0 = blockIdx.x * 256;
    const int tid    = chain0 + tidb;
    const int f      = tid & (FEAT - 1);

    // per-feature parameters -> registers
    const float Wdgx = w_dg_x[f], Wdgh = w_dg_h[f];
    const float Wxz = w_xz[f], Whz = w_hz[f], Wmz = w_mz[f];
    const float Wxr = w_xr[f], Whr = w_hr[f], Wmr = w_mr[f];
    const float Wxh = w_xh[f], Whh = w_hh[f], Wmh = w_mh[f];
    const float Bdgx = b_dg_x[f], Bdgh = b_dg_h[f];
    const float Bz = b_z[f], Br = b_r[f], Bh = b_h[f];
    const float Xmean = x_mean[f];

    const int ntiles = TLEN / TT;                // 256 tiles

    // prologue: kick off tile 0 into buffer 0 (one wave issues; EXEC ignored by TDM)
    if (tidb == 0) {
        const size_t off0 = (size_t)chain0 * TLEN;
        tdm_load_2d((unsigned)(uintptr_t)&sX[0][0], X     + off0);
        tdm_load_2d((unsigned)(uintptr_t)&sM[0][0], Mask  + off0);
        tdm_load_2d((unsigned)(uintptr_t)&sD[0][0], Delta + off0);
    }

    float h = 0.0f;

    for (int it = 0; it < ntiles; ++it) {
        const int cur = it & 1;

        if (tidb == 0) {
            if (it + 1 < ntiles) {
                const int nxt = cur ^ 1;
                const size_t off = (size_t)chain0 * TLEN + (size_t)(it + 1) * TT;
                tdm_load_2d((unsigned)(uintptr_t)&sX[nxt][0], X     + off);
                tdm_load_2d((unsigned)(uintptr_t)&sM[nxt][0], Mask  + off);
                tdm_load_2d((unsigned)(uintptr_t)&sD[nxt][0], Delta + off);
                // TDM ops complete in order: <=3 outstanding => tile `it` landed
                __builtin_amdgcn_s_wait_tensorcnt(3);
            } else {
                __builtin_amdgcn_s_wait_tensorcnt(0);
            }
        }
        __syncthreads();   // tile `it` visible to all waves

        const float* __restrict__ xr = &sX[cur][tidb * RSTR];
        const float* __restrict__ mr = &sM[cur][tidb * RSTR];
        const float* __restrict__ dr = &sD[cur][tidb * RSTR];

#pragma unroll
        for (int j = 0; j < TT; ++j) {
            const float x0 = xr[j], m = mr[j], d = dr[j];

            const float gx = fexp(-fmaxf(0.0f, __builtin_fmaf(Wdgx, d, Bdgx)));
            const float gh = fexp(-fmaxf(0.0f, __builtin_fmaf(Wdgh, d, Bdgh)));

            const float ximp = __builtin_fmaf(gx, x0, (1.0f - gx) * Xmean);
            const float x    = __builtin_fmaf(m, x0, (1.0f - m) * ximp);

            h = gh * h;

            const float z  = fsigmoid(Wxz * x + Whz * h + Wmz * m + Bz);
            const float r  = fsigmoid(Wxr * x + Whr * h + Wmr * m + Br);
            const float ht = ftanh(Wxh * x + Whh * (r * h) + Wmh * m + Bh);

            h = __builtin_fmaf(1.0f - z, h, z * ht);
        }

        __syncthreads();   // compute on `cur` done before tile it+2 overwrites it
    }

    hout[tid] = h;   // [BATCH][FEAT] row-major
}

// ============================================================
// Kernel 2: out(128x2) = h(128x256) @ w_hy(256x2) + b_y
// via V_WMMA_F32_16X16X4_F32 (exact fp32 matrix pipe).
// One wave per 16-row M tile; N padded 2->16 with zeros; K=256 in 64 steps.
// ============================================================
__global__ __launch_bounds__(256) void grud_proj_kernel(
    const float* __restrict__ h,      // [BATCH][FEAT]
    const float* __restrict__ w_hy,   // [FEAT][NOUT]
    const float* __restrict__ b_y,    // [NOUT]
    float* __restrict__ out)          // [BATCH][NOUT]
{
    const int lane    = threadIdx.x & 31;
    const int wave    = threadIdx.x >> 5;   // 0..7 -> M tile
    const int m0      = wave * 16;
    const int halfSel = lane >> 4;          // 0: K pair {0,1}, 1: K pair {2,3}
    const int lrow    = lane & 15;          // A: row M ; B: column N

    v8f c = {};

#pragma unroll 4
    for (int k0 = 0; k0 < FEAT; k0 += 4) {
        const float* ap = h + (size_t)(m0 + lrow) * FEAT + k0 + halfSel * 2;
        v2f a;
        a.x = ap[0];
        a.y = ap[1];

        v2f b;
        b.x = 0.0f;
        b.y = 0.0f;
        if (lrow < NOUT) {
            b.x = w_hy[(k0 + halfSel * 2 + 0) * NOUT + lrow];
            b.y = w_hy[(k0 + halfSel * 2 + 1) * NOUT + lrow];
        }

        c = __builtin_amdgcn_wmma_f32_16x16x4_f32(
                false, a, false, b, (short)0, c, false, false);
    }

    if (lrow < NOUT) {
        const float bias = b_y[lrow];
#pragma unroll
        for (int j = 0; j < 8; ++j) {
            const int M = m0 + j + 8 * halfSel;
            out[M * NOUT + lrow] = c[j] + bias;
        }
    }
}

// ============================================================
extern "C" void kernel_launch(void* const* d_in, const int* in_sizes, int n_in,
                              void* d_out, int out_size, void* d_ws, size_t ws_size,
                              hipStream_t stream) {
    const float* X      = (const float*)d_in[0];
    const float* Mask   = (const float*)d_in[1];
    const float* Delta  = (const float*)d_in[2];
    const float* x_mean = (const float*)d_in[3];
    const float* w_dg_x = (const float*)d_in[4];
    const float* w_dg_h = (const float*)d_in[5];
    const float* w_xz   = (const float*)d_in[6];
    const float* w_hz   = (const float*)d_in[7];
    const float* w_mz   = (const float*)d_in[8];
    const float* w_xr   = (const float*)d_in[9];
    const float* w_hr   = (const float*)d_in[10];
    const float* w_mr   = (const float*)d_in[11];
    const float* w_xh   = (const float*)d_in[12];
    const float* w_hh   = (const float*)d_in[13];
    const float* w_mh   = (const float*)d_in[14];
    const float* w_hy   = (const float*)d_in[15];
    const float* b_dg_x = (const float*)d_in[16];
    const float* b_dg_h = (const float*)d_in[17];
    const float* b_z    = (const float*)d_in[18];
    const float* b_r    = (const float*)d_in[19];
    const float* b_h    = (const float*)d_in[20];
    const float* b_y    = (const float*)d_in[21];

    float* hbuf = (float*)d_ws;            // BATCH*FEAT floats = 128 KB
    float* out  = (float*)d_out;           // BATCH*NOUT floats

    const int nchains = BATCH * FEAT;      // 32768 chains, 128 blocks
    grud_scan_kernel<<<nchains / 256, 256, 0, stream>>>(
        X, Mask, Delta, x_mean,
        w_dg_x, w_dg_h, w_xz, w_hz, w_mz, w_xr, w_hr, w_mr,
        w_xh, w_hh, w_mh, b_dg_x, b_dg_h, b_z, b_r, b_h, hbuf);

    grud_proj_kernel<<<1, 256, 0, stream>>>(hbuf, w_hy, b_y, out);
}